// ConvPolicy30_20761871909588
// MI455X (gfx1250) — compile-verified
//
#include <hip/hip_runtime.h>
#include <math.h>

typedef __attribute__((ext_vector_type(2))) float v2f;
typedef __attribute__((ext_vector_type(8))) float v8f;

#if __has_builtin(__builtin_amdgcn_wmma_f32_16x16x4_f32)
#define HAVE_WMMA_F32X4 1
#else
#define HAVE_WMMA_F32X4 0
#endif

// D(16x16,f32) += A(16x4) * B(4x16), K-chunked.
// Aw: 16 rows (M) x 48 cols (u), zero-padded. Bw: 16 rows (N) x 48 cols (u),
// zero-padded (B stored transposed so both operands load identically).
// ISA 7.12.2: lanes 0-15 hold K={0,1} in the VGPR pair, lanes 16-31 K={2,3}.
template <int NCH>
__device__ __forceinline__ v8f wmma_mm_f32(const float* __restrict__ Aw,
                                           const float* __restrict__ Bw,
                                           int tid) {
  v8f c = {0.f, 0.f, 0.f, 0.f, 0.f, 0.f, 0.f, 0.f};
  const int base = (tid & 15) * 48 + ((tid >> 4) << 1);  // even -> 8B aligned
#if HAVE_WMMA_F32X4
#pragma unroll
  for (int k4 = 0; k4 < NCH; ++k4) {
    const v2f a = *(const v2f*)(Aw + base + 4 * k4);
    const v2f b = *(const v2f*)(Bw + base + 4 * k4);
    c = __builtin_amdgcn_wmma_f32_16x16x4_f32(false, a, false, b, (short)0, c,
                                              false, false);
  }
#else
  const int n = tid & 15;
  if ((tid >> 4) == 0) {
    for (int r = 0; r < 8; ++r) {
      float s = 0.f;
      for (int u = 0; u < 4 * NCH; ++u) s += Aw[r * 48 + u] * Bw[n * 48 + u];
      c[r] = s;
    }
  }
#endif
  return c;
}

// ConvTranspose1d (stride 1, k=3): y[oc][p] = sum_{ic,t} w[ic][oc][2-t] * x[ic][p+t-(2-pad)]
__device__ __forceinline__ float convT_out(const float* __restrict__ xin, int Lin,
                                           int IC, int OC,
                                           const float* __restrict__ w, int pad,
                                           int oc, int p) {
  float acc = 0.f;
  const int padp = 2 - pad;
  for (int ic = 0; ic < IC; ++ic)
    for (int t = 0; t < 3; ++t) {
      const int j = p + t - padp;
      if (j >= 0 && j < Lin)
        acc += w[(ic * OC + oc) * 3 + (2 - t)] * xin[ic * Lin + j];
    }
  return acc;
}

__global__ __launch_bounds__(32) void conv_policy_kernel(
    const float* __restrict__ x,
    const float* __restrict__ w1, const float* __restrict__ b1,
    const float* __restrict__ w2, const float* __restrict__ b2,
    const float* __restrict__ w3, const float* __restrict__ b3,
    const float* __restrict__ ew1, const float* __restrict__ eb1,
    const float* __restrict__ ew2, const float* __restrict__ eb2,
    const float* __restrict__ dw1, const float* __restrict__ db1,
    const float* __restrict__ dw2, const float* __restrict__ db2,
    const float* __restrict__ dw3, const float* __restrict__ db3,
    const float* __restrict__ dw4, const float* __restrict__ db4,
    float* __restrict__ out) {
  const int tid = threadIdx.x;  // 0..31, one wave32

  __shared__ float jcat[12 * 15];               // encoder input (6 jl + 6 jdl)
  __shared__ __align__(16) float Aw[16 * 48];   // WMMA A (weights), zero-padded
  __shared__ __align__(16) float Bw[16 * 48];   // WMMA B^T (im2col), zero-padded
  __shared__ float c1[6 * 13];                  // conv1 out
  __shared__ float dsb[6 * 5];                  // adaptive-pool out
  __shared__ float c2[4 * 3];                   // conv2 out
  __shared__ float embin[6];                    // [conv3(4), psi, x[100]]
  __shared__ float e1[4], e2[4];
  __shared__ float d1[4 * 3];                   // dec1 out
  __shared__ float d2[2 * 5];                   // dec2 out
  __shared__ float d2u[2 * 13];                 // upsample out
  __shared__ float d3[2 * 15];                  // dec3 out
  __shared__ float o4[6 * 15];                  // dec4 out

  // ---- stage jcat: [zeros(2), x[7:95]] and [zeros(2), x[101:189]] as (6,15)+(6,15)
  for (int i = tid; i < 180; i += 32) {
    const int ch = i / 15, l = i % 15;
    float v;
    if (ch < 6) {
      const int f = ch * 15 + l;
      v = (f < 2) ? 0.f : x[7 + f - 2];
    } else {
      const int f = (ch - 6) * 15 + l;
      v = (f < 2) ? 0.f : x[101 + f - 2];
    }
    jcat[i] = v;
  }
  for (int i = tid; i < 16 * 48; i += 32) Aw[i] = 0.f;
  for (int i = tid; i < 16 * 48; i += 32) Bw[i] = 0.f;
  __syncthreads();

  // conv1 weights (6,12,3): Aw[m][u] = w1[m*36+u], u = ic*3+t
  for (int i = tid; i < 6 * 36; i += 32) Aw[(i / 36) * 48 + (i % 36)] = w1[i];
  // conv1 im2col: Bw[n][u] = jcat[(u/3)*15 + n + u%3], n<13, u<36
  for (int i = tid; i < 13 * 36; i += 32) {
    const int n = i % 13, u = i / 13;
    Bw[n * 48 + u] = jcat[(u / 3) * 15 + n + (u % 3)];
  }
  __syncthreads();

  // ---- conv1 via WMMA f32 16x16x4: M=6(oc), N=13(pos), K=36
  v8f acc1 = wmma_mm_f32<9>(Aw, Bw, tid);
  {
    const int n = tid & 15, half = tid >> 4;
    if (half == 0 && n < 13)
      for (int r = 0; r < 6; ++r) c1[r * 13 + n] = tanhf(acc1[r] + b1[r]);
  }
  __syncthreads();

  // ---- adaptive avg-pool 13 -> 5
  if (tid < 30) {
    const int ch = tid / 5, o = tid % 5;
    const int s = (o * 13) / 5, e = ((o + 1) * 13 + 4) / 5;
    float a = 0.f;
    for (int l = s; l < e; ++l) a += c1[ch * 13 + l];
    dsb[ch * 5 + o] = a / (float)(e - s);
  }
  __syncthreads();

  // ---- conv2 (6->4, k3) -> (4,3)
  if (tid < 12) {
    const int oc = tid / 3, p = tid % 3;
    float a = b2[oc];
    for (int ic = 0; ic < 6; ++ic)
      for (int t = 0; t < 3; ++t)
        a += w2[oc * 18 + ic * 3 + t] * dsb[ic * 5 + p + t];
    c2[oc * 3 + p] = tanhf(a);
  }
  __syncthreads();

  // ---- conv3 (4->4, k3) -> (4,1); mean over len-1 axis is identity
  if (tid < 4) {
    float a = b3[tid];
    for (int ic = 0; ic < 4; ++ic)
      for (int t = 0; t < 3; ++t) a += w3[tid * 12 + ic * 3 + t] * c2[ic * 3 + t];
    embin[tid] = tanhf(a);
  }
  if (tid == 0) {
    const float qw = x[3], qx = x[4], qy = x[5], qz = x[6];
    embin[4] = atan2f(qz, qw) - atan2f(-qx, qy);  // psi
    embin[5] = x[100];                            // obsd[:, -1]
  }
  __syncthreads();

  // ---- emb1 (6->4, k1), emb2 (4->4, k1)
  if (tid < 4) {
    float a = eb1[tid];
    for (int ic = 0; ic < 6; ++ic) a += ew1[tid * 6 + ic] * embin[ic];
    e1[tid] = tanhf(a);
  }
  __syncthreads();
  if (tid < 4) {
    float a = eb2[tid];
    for (int ic = 0; ic < 4; ++ic) a += ew2[tid * 4 + ic] * e1[ic];
    e2[tid] = tanhf(a);
  }
  __syncthreads();

  // ---- dec1: convT (4->4), len 1 -> 3
  if (tid < 12) {
    const int oc = tid / 3, p = tid % 3;
    d1[tid] = tanhf(convT_out(e2, 1, 4, 4, dw1, 0, oc, p) + db1[oc]);
  }
  __syncthreads();
  // ---- dec2: convT (4->2), len 3 -> 5
  if (tid < 10) {
    const int oc = tid / 5, p = tid % 5;
    d2[tid] = tanhf(convT_out(d1, 3, 4, 2, dw2, 0, oc, p) + db2[oc]);
  }
  __syncthreads();
  // ---- nearest upsample 5 -> 13: idx = floor(i*5/13)
  if (tid < 26) {
    const int ch = tid / 13, i = tid % 13;
    d2u[tid] = d2[ch * 5 + (i * 5) / 13];
  }
  __syncthreads();
  // ---- dec3: convT (2->2), len 13 -> 15
  if (tid < 30) {
    const int oc = tid / 15, p = tid % 15;
    d3[tid] = tanhf(convT_out(d2u, 13, 2, 2, dw3, 0, oc, p) + db3[oc]);
  }
  __syncthreads();

  // ---- dec4 staging. Fill regions are supersets of conv1's fills, and the
  // zero-padding (u in [42,48), n in [15,16)) was never written -> no re-zero.
  // A: wc[oc][ic][t] = dw4[ic][oc][2-t]; u = ic*3+t, m<6, u<42
  for (int i = tid; i < 6 * 42; i += 32) {
    const int m = i / 42, u = i % 42;
    const int ic = u / 3, t = u % 3;
    Aw[m * 48 + u] = dw4[ic * 18 + m * 3 + (2 - t)];
  }
  // B^T: input = concat([d3(2ch), jcat(12ch)]), ConvT pad'=1 applied inline.
  for (int i = tid; i < 15 * 42; i += 32) {
    const int n = i % 15, u = i / 15;
    const int ch = u / 3, t = u % 3;
    const int j = n + t;  // position in 1-padded row [0,17)
    float v = 0.f;
    if (j >= 1 && j <= 15) {
      const int l = j - 1;
      v = (ch < 2) ? d3[ch * 15 + l] : jcat[(ch - 2) * 15 + l];
    }
    Bw[n * 48 + u] = v;
  }
  __syncthreads();

  // ---- dec4 via WMMA: M=6(oc), N=15(pos), K=42 (padded to 44)
  v8f acc4 = wmma_mm_f32<11>(Aw, Bw, tid);
  {
    const int n = tid & 15, half = tid >> 4;
    if (half == 0 && n < 15)
      for (int r = 0; r < 6; ++r) o4[r * 15 + n] = acc4[r] + db4[r];
  }
  __syncthreads();

  // ---- output: flatten (6,15) -> 90, drop first 2 -> 88
  for (int i = tid; i < 88; i += 32) out[i] = o4[i + 2];
}

extern "C" void kernel_launch(void* const* d_in, const int* in_sizes, int n_in,
                              void* d_out, int out_size, void* d_ws, size_t ws_size,
                              hipStream_t stream) {
  (void)in_sizes; (void)n_in; (void)out_size; (void)d_ws; (void)ws_size;
  const float* x   = (const float*)d_in[0];
  const float* w1  = (const float*)d_in[1];
  const float* b1  = (const float*)d_in[2];
  const float* w2  = (const float*)d_in[3];
  const float* b2  = (const float*)d_in[4];
  const float* w3  = (const float*)d_in[5];
  const float* b3  = (const float*)d_in[6];
  const float* ew1 = (const float*)d_in[7];
  const float* eb1 = (const float*)d_in[8];
  const float* ew2 = (const float*)d_in[9];
  const float* eb2 = (const float*)d_in[10];
  const float* dw1 = (const float*)d_in[11];
  const float* db1 = (const float*)d_in[12];
  const float* dw2 = (const float*)d_in[13];
  const float* db2 = (const float*)d_in[14];
  const float* dw3 = (const float*)d_in[15];
  const float* db3 = (const float*)d_in[16];
  const float* dw4 = (const float*)d_in[17];
  const float* db4 = (const float*)d_in[18];
  float* out = (float*)d_out;

  conv_policy_kernel<<<1, 32, 0, stream>>>(
      x, w1, b1, w2, b2, w3, b3, ew1, eb1, ew2, eb2,
      dw1, db1, dw2, db2, dw3, db3, dw4, db4, out);
}